// InputLayer_83880711291486
// MI455X (gfx1250) — compile-verified
//
#include <hip/hip_runtime.h>
#include <stdint.h>

#define VOCAB  50257
#define DMODEL 1024
#define NTOK   (4 * 4096)

typedef float v4f __attribute__((ext_vector_type(4)));

// ---------------------------------------------------------------------------
// gfx1250 async global->LDS copy (ASYNCcnt-tracked DMA, bypasses VGPRs).
// vdst = LDS byte address, vaddr = 64-bit global address (GV mode).
// TH_LOAD_NT: W is streamed exactly once -> don't evict WT from L2.
// ---------------------------------------------------------------------------
__device__ __forceinline__ void async_load_to_lds_b32_nt(uint32_t lds_byte_off,
                                                         const float* gaddr) {
    asm volatile("global_load_async_to_lds_b32 %0, %1, off th:TH_LOAD_NT"
                 :: "v"(lds_byte_off), "v"(gaddr)
                 : "memory");
}

__device__ __forceinline__ void wait_asynccnt0() {
    asm volatile("s_wait_asynccnt 0" ::: "memory");
}

// ---------------------------------------------------------------------------
// Pass 1: transpose W [DMODEL, VOCAB] -> WT [VOCAB, DMODEL] via LDS tiles.
// The async DMA deposits each element already transposed in LDS, so both the
// global read (along vocab) and the global write (along d_model) are fully
// coalesced. Tile pitch 33 dwords -> bank-conflict-free.
// Block: 32 (x: vocab lane) x 8 (y: d row), 4 rows per thread.
// ---------------------------------------------------------------------------
__global__ __launch_bounds__(256)
void transpose_kernel(const float* __restrict__ W, float* __restrict__ WT) {
    __shared__ float tile[32][33];   // tile[v_local][d_local]

    const int tx = threadIdx.x;          // 0..31
    const int ty = threadIdx.y;          // 0..7
    const int v0 = blockIdx.x * 32;      // vocab tile origin
    const int d0 = blockIdx.y * 32;      // d_model tile origin

    // Clamp instead of branching so EXEC stays full for the async ops.
    const int v  = v0 + tx;
    const int vc = (v < VOCAB) ? v : (VOCAB - 1);

#pragma unroll
    for (int k = 0; k < 4; ++k) {
        const int d = d0 + ty + 8 * k;                       // < 1024 always
        const uint32_t lds_off =
            (uint32_t)(uintptr_t)&tile[tx][ty + 8 * k];      // transposed slot
        async_load_to_lds_b32_nt(lds_off, W + (size_t)d * VOCAB + vc);
    }

    wait_asynccnt0();     // our wave's DMA deposits are visible in LDS
    __syncthreads();      // cross-wave visibility within the workgroup

#pragma unroll
    for (int k = 0; k < 4; ++k) {
        const int vrow = v0 + ty + 8 * k;
        if (vrow < VOCAB) {
            // Default (RT) stores: WT is re-read in pass 2, keep it cacheable.
            WT[(size_t)vrow * DMODEL + (d0 + tx)] = tile[ty + 8 * k][tx];
        }
    }
}

// ---------------------------------------------------------------------------
// Pass 2: out[token, :] = WT[ids[token], :]   (4 KB contiguous row copy)
// One block per token; 256 threads x float4 = 1024 floats. B128 coalesced.
// Output is write-once -> non-temporal stores keep WT lines resident in L2.
// ---------------------------------------------------------------------------
__global__ __launch_bounds__(256)
void gather_kernel(const int* __restrict__ ids,
                   const float* __restrict__ WT,
                   float* __restrict__ out) {
    const int token = blockIdx.x;
    const int id    = ids[token];        // uniform per block -> scalar load
    const v4f* __restrict__ src = (const v4f*)(WT + (size_t)id * DMODEL);
    v4f*       __restrict__ dst = (v4f*)(out + (size_t)token * DMODEL);
    const v4f val = src[threadIdx.x];                 // RT: may hit L2 (WT)
    __builtin_nontemporal_store(val, &dst[threadIdx.x]);
}

// ---------------------------------------------------------------------------
// Fallback (workspace too small): direct strided gather from W.
// Lanes map to d_model so the 64 MiB of stores stay coalesced (and NT);
// reads lean on the 192 MB L2 (W is ~206 MB, lines reused ~10x over tokens).
// ---------------------------------------------------------------------------
__global__ __launch_bounds__(256)
void direct_gather_kernel(const int* __restrict__ ids,
                          const float* __restrict__ W,
                          float* __restrict__ out) {
    const int token = blockIdx.x;
    const int id    = ids[token];
#pragma unroll
    for (int k = 0; k < 4; ++k) {
        const int d = threadIdx.x + 256 * k;
        const float val = W[(size_t)d * VOCAB + id];
        __builtin_nontemporal_store(val, &out[(size_t)token * DMODEL + d]);
    }
}

// ---------------------------------------------------------------------------
extern "C" void kernel_launch(void* const* d_in, const int* in_sizes, int n_in,
                              void* d_out, int out_size, void* d_ws,
                              size_t ws_size, hipStream_t stream) {
    (void)in_sizes; (void)n_in; (void)out_size;

    const int*   ids = (const int*)d_in[0];     // [4, 4096] int32
    const float* W   = (const float*)d_in[1];   // [1024, 50257] f32
    float*       out = (float*)d_out;           // [4, 4096, 1024] f32

    const size_t need = (size_t)VOCAB * DMODEL * sizeof(float);  // ~196 MiB

    if (d_ws != nullptr && ws_size >= need) {
        float* WT = (float*)d_ws;
        dim3 tgrid((VOCAB + 31) / 32, DMODEL / 32);
        transpose_kernel<<<tgrid, dim3(32, 8), 0, stream>>>(W, WT);
        gather_kernel<<<NTOK, 256, 0, stream>>>(ids, WT, out);
    } else {
        direct_gather_kernel<<<NTOK, 256, 0, stream>>>(ids, W, out);
    }
}